// T5DecoderSelfAttention_88536455839770
// MI455X (gfx1250) — compile-verified
//
#include <hip/hip_runtime.h>
#include <hip/hip_bf16.h>
#include <math.h>

// Problem constants (from reference)
#define BB   16
#define SS   1
#define HID  2048
#define HH   32
#define DD   64
#define KVN  4096
#define SCALE 0.125f   // 64^-0.5

typedef float v2f __attribute__((ext_vector_type(2)));
typedef float v8f __attribute__((ext_vector_type(8)));

// ---------------------------------------------------------------------------
// Kernel 1/3: C(16 x N) = A(16 x K, row-major) * Bw(N x K, row-major)^T
// One wave computes one 16x16 output tile, looping K in steps of 4 with
// V_WMMA_F32_16X16X4_F32.
//
// f32 WMMA VGPR layouts (CDNA5 ISA 7.12.2):
//   A (16x4):  lane l (m = l&15, half = l>>4) holds {A[m][2*half], A[m][2*half+1]}
//   B (4x16):  lane l (n = l&15, half = l>>4) holds {B[2*half][n],  B[2*half+1][n]}
//              and since B = Bw^T, B[k][n] = Bw[n0+n][k]  -> contiguous float2
//   C (16x16): VGPR r in lane l holds C[r + 8*half][n]
// ---------------------------------------------------------------------------
__global__ __launch_bounds__(256)
void gemm16_wmma(const float* __restrict__ A,   // 16 x K
                 const float* __restrict__ Bw,  // N x K
                 float* __restrict__ Cout,      // 16 x N
                 int K, int N) {
    const int wave = threadIdx.x >> 5;
    const int lane = threadIdx.x & 31;
    const int n0   = (blockIdx.x * 8 + wave) * 16;
    if (n0 >= N) return;                 // wave-uniform: EXEC stays all-ones

    const int half = lane >> 4;          // 0 or 1
    const int mn   = lane & 15;          // row of A / col of C, row of Bw

    v8f c = {0.f, 0.f, 0.f, 0.f, 0.f, 0.f, 0.f, 0.f};

    const float* arow = A  + (size_t)mn * K + 2 * half;
    const float* brow = Bw + (size_t)(n0 + mn) * K + 2 * half;

#if __has_builtin(__builtin_amdgcn_wmma_f32_16x16x4_f32)
    #pragma unroll 8
    for (int k = 0; k < K; k += 4) {
        v2f av = *(const v2f*)(arow + k);
        v2f bv = *(const v2f*)(brow + k);
        // (neg_a, A, neg_b, B, c_mod, C, reuse_a, reuse_b)
        c = __builtin_amdgcn_wmma_f32_16x16x4_f32(
                false, av, false, bv, (short)0, c, false, false);
    }
    #pragma unroll
    for (int r = 0; r < 8; ++r) {
        const int m = r + 8 * half;
        Cout[(size_t)m * N + n0 + mn] = c[r];
    }
#else
    // Fallback (compile-safety only): direct dot products, same tile mapping.
    #pragma unroll
    for (int r = 0; r < 8; ++r) {
        const int m = r + 8 * half;
        float acc = 0.f;
        const float* ar = A  + (size_t)m * K;
        const float* br = Bw + (size_t)(n0 + mn) * K;
        for (int k = 0; k < K; ++k) acc = fmaf(ar[k], br[k], acc);
        Cout[(size_t)m * N + n0 + mn] = acc;
    }
#endif
}

// ---------------------------------------------------------------------------
// Kernel 2: fused single-query attention per (b,h). 512 blocks x 256 threads.
//   scores[j] = SCALE * dot(q, k_j) + bias[b,h,0,j]   (k_4095 = k_new)
//   probs     = softmax(scores)
//   a[b, h*64+d] = sum_j probs[j] * v_j[d]            (v_4095 = v_new)
// Streams 2 MB of K/V per block; this kernel is the ~1.07 GB (512 x 2 MiB) floor.
// ---------------------------------------------------------------------------
__global__ __launch_bounds__(256)
void attention_kernel(const float* __restrict__ qkv,      // 16 x 6144
                      const float* __restrict__ bias,     // (B*H) x KV
                      const float* __restrict__ past_kv,  // 2 x B*H x KV x D
                      float* __restrict__ a_out) {        // 16 x 2048
    __shared__ float s_q[DD];
    __shared__ float s_scores[KVN];      // 16 KB
    __shared__ float s_red[256];
    __shared__ float s_acc[8 * DD];      // per-wave partial outputs

    const int tid = threadIdx.x;
    const int bh  = blockIdx.x;          // b*H + h
    const int b   = bh >> 5;
    const int h   = bh & 31;

    const float* qp   = qkv + (size_t)b * (3 * HID) + h * DD;                 // q
    const float* knew = qkv + (size_t)b * (3 * HID) + HID + h * DD;           // k_new
    const float* vnew = qkv + (size_t)b * (3 * HID) + 2 * HID + h * DD;       // v_new

    const size_t head_off = (size_t)bh * KVN * DD;
    const float* Kbase = past_kv + head_off;
    const float* Vbase = past_kv + (size_t)BB * HH * KVN * DD + head_off;
    const float* biasp = bias + (size_t)bh * KVN;

    if (tid < DD) s_q[tid] = qp[tid];
    __syncthreads();

    // ---- Phase 1: scores -------------------------------------------------
    const float4* q4 = (const float4*)s_q;
    for (int j = tid; j < KVN; j += 256) {
        const float*  krow = (j == KVN - 1) ? knew : (Kbase + (size_t)j * DD);
        const float4* k4   = (const float4*)krow;
        float acc = 0.f;
        #pragma unroll
        for (int i = 0; i < DD / 4; ++i) {
            float4 kk = k4[i];
            float4 qq = q4[i];
            acc = fmaf(qq.x, kk.x, acc);
            acc = fmaf(qq.y, kk.y, acc);
            acc = fmaf(qq.z, kk.z, acc);
            acc = fmaf(qq.w, kk.w, acc);
        }
        s_scores[j] = fmaf(acc, SCALE, biasp[j]);
    }
    __syncthreads();

    // ---- Phase 2: softmax ------------------------------------------------
    float lm = -INFINITY;
    for (int j = tid; j < KVN; j += 256) lm = fmaxf(lm, s_scores[j]);
    s_red[tid] = lm;
    __syncthreads();
    for (int s = 128; s > 0; s >>= 1) {
        if (tid < s) s_red[tid] = fmaxf(s_red[tid], s_red[tid + s]);
        __syncthreads();
    }
    const float gmax = s_red[0];
    __syncthreads();

    float lsum = 0.f;
    for (int j = tid; j < KVN; j += 256) {
        float e = __expf(s_scores[j] - gmax);
        s_scores[j] = e;
        lsum += e;
    }
    s_red[tid] = lsum;
    __syncthreads();
    for (int s = 128; s > 0; s >>= 1) {
        if (tid < s) s_red[tid] += s_red[tid + s];
        __syncthreads();
    }
    const float inv = 1.0f / s_red[0];
    __syncthreads();

    // ---- Phase 3: out = probs . V  (wave-per-row, lane-per-2-dims) ------
    const int wave = tid >> 5;
    const int lane = tid & 31;
    float ax = 0.f, ay = 0.f;
    for (int j = wave; j < KVN; j += 8) {
        const float  p    = s_scores[j];
        const float* vrow = (j == KVN - 1) ? vnew : (Vbase + (size_t)j * DD);
        const float2 vv   = ((const float2*)vrow)[lane];
        ax = fmaf(p, vv.x, ax);
        ay = fmaf(p, vv.y, ay);
    }
    s_acc[wave * DD + 2 * lane]     = ax;
    s_acc[wave * DD + 2 * lane + 1] = ay;
    __syncthreads();

    if (tid < DD) {
        float r = 0.f;
        #pragma unroll
        for (int w = 0; w < 8; ++w) r += s_acc[w * DD + tid];
        a_out[(size_t)b * HID + h * DD + tid] = r * inv;
    }
}

// ---------------------------------------------------------------------------
extern "C" void kernel_launch(void* const* d_in, const int* in_sizes, int n_in,
                              void* d_out, int out_size, void* d_ws, size_t ws_size,
                              hipStream_t stream) {
    const float* x       = (const float*)d_in[0];  // 16 x 1 x 2048
    const float* bias    = (const float*)d_in[1];  // 16 x 32 x 1 x 4096
    const float* past_kv = (const float*)d_in[2];  // 2 x 16 x 32 x 4096 x 64
    const float* w_qkv   = (const float*)d_in[3];  // 6144 x 2048
    const float* w_o     = (const float*)d_in[4];  // 2048 x 2048
    float*       out     = (float*)d_out;          // 16 x 1 x 2048

    float* ws_qkv = (float*)d_ws;                              // 16 x 6144
    float* ws_a   = ws_qkv + (size_t)BB * 3 * HID;             // 16 x 2048

    // 1) QKV projection: (16 x 2048) * (6144 x 2048)^T -> 16 x 6144
    //    384 16x16 tiles -> 48 blocks of 8 waves
    gemm16_wmma<<<48, 256, 0, stream>>>(x, w_qkv, ws_qkv, HID, 3 * HID);

    // 2) Fused attention: one block per (b,h)
    attention_kernel<<<BB * HH, 256, 0, stream>>>(ws_qkv, bias, past_kv, ws_a);

    // 3) Output projection: (16 x 2048) * (2048 x 2048)^T -> 16 x 2048
    //    128 tiles -> 16 blocks of 8 waves
    gemm16_wmma<<<16, 256, 0, stream>>>(ws_a, w_o, out, HID, HID);
}